// BiGCNEncoder_36120674959506
// MI455X (gfx1250) — compile-verified
//
#include <hip/hip_runtime.h>
#include <stdint.h>

#define D 32
#define NWIN 8
#define BN_EPS 1e-5f

typedef __attribute__((ext_vector_type(2))) float v2f;
typedef __attribute__((ext_vector_type(8))) float v8f;

// ---------------------------------------------------------------------------
// zero-fill (uint4 vectorized; all call sites pass counts divisible by 4)
// ---------------------------------------------------------------------------
__global__ void fill_zero_k(uint32_t* __restrict__ p, long n4) {
    long i = (long)blockIdx.x * blockDim.x + threadIdx.x;
    if (i < n4) {
        uint4 z = {0u, 0u, 0u, 0u};
        ((uint4*)p)[i] = z;
    }
}

// ---------------------------------------------------------------------------
// per-window in-degree counting + bucket histogram
// ---------------------------------------------------------------------------
__global__ void count_edges_k(const int* __restrict__ dst, const int* __restrict__ ewin,
                              int* __restrict__ cntw, int* __restrict__ bktcnt,
                              int E, int N) {
    int e = blockIdx.x * blockDim.x + threadIdx.x;
    if (e < E) {
        int w = ewin[e];
        atomicAdd(&cntw[(size_t)w * N + dst[e]], 1);
        atomicAdd(&bktcnt[w], 1);
    }
}

// deg_w = cnt_w + 1 (self loop); deg_full = sum_w cnt_w + 1
__global__ void deg_finalize_k(const int* __restrict__ cntw,
                               float* __restrict__ rdegw, float* __restrict__ invw,
                               float* __restrict__ rdegf, float* __restrict__ invf,
                               int N) {
    int n = blockIdx.x * blockDim.x + threadIdx.x;
    if (n >= N) return;
    int tot = 0;
#pragma unroll
    for (int w = 0; w < NWIN; ++w) {
        int c = cntw[(size_t)w * N + n];
        tot += c;
        float dg = (float)(c + 1);
        rdegw[(size_t)w * N + n] = rsqrtf(dg);
        invw [(size_t)w * N + n] = 1.0f / dg;
    }
    float dgf = (float)(tot + 1);
    rdegf[n] = rsqrtf(dgf);
    invf [n] = 1.0f / dgf;
}

__global__ void bucket_prefix_k(const int* __restrict__ bktcnt,
                                int* __restrict__ bktoff, int* __restrict__ cursor) {
    if (blockIdx.x == 0 && threadIdx.x == 0) {
        int s = 0;
        for (int w = 0; w < NWIN; ++w) { bktoff[w] = s; s += bktcnt[w]; cursor[w] = 0; }
    }
}

__global__ void bucket_scatter_k(const int* __restrict__ ewin, const int* __restrict__ bktoff,
                                 int* __restrict__ cursor, int* __restrict__ ebkt, int E) {
    int e = blockIdx.x * blockDim.x + threadIdx.x;
    if (e < E) {
        int w = ewin[e];
        int pos = bktoff[w] + atomicAdd(&cursor[w], 1);
        ebkt[pos] = e;
    }
}

// ---------------------------------------------------------------------------
// GEMM: C[nrows,32] = A[nrows,32] @ W[32,32] (+bias) via V_WMMA_F32_16X16X4_F32
// one wave -> one 16-row tile; W staged in LDS, B fragments via ds_load
// A-frag (16x4 f32): lane L holds M=L%16, K = 4k + 2*(L/16) + {0,1}
// C-frag (16x16 f32): VGPR r -> M = r + 8*(L/16), N = L%16
// ---------------------------------------------------------------------------
__global__ void gemm32_k(const float* __restrict__ A, const float* __restrict__ W,
                         const float* __restrict__ bias, int has_bias,
                         float* __restrict__ C, int nrows) {
    __shared__ float sW[D * D];
    for (int i = threadIdx.x; i < D * D; i += blockDim.x) sW[i] = W[i];
    __syncthreads();

    int wave = threadIdx.x >> 5;
    int lane = threadIdx.x & 31;
    int row0 = (blockIdx.x * 4 + wave) * 16;
    if (row0 >= nrows) return;          // uniform per wave (nrows % 16 == 0)

    int m = lane & 15;                  // M / N index within tile
    int h = lane >> 4;                  // half-wave selector

    v2f a[8];
    const float* arow = A + (size_t)(row0 + m) * D;
#pragma unroll
    for (int k = 0; k < 8; ++k)
        a[k] = *(const v2f*)(arow + 4 * k + 2 * h);

#pragma unroll
    for (int nt = 0; nt < 2; ++nt) {
        v8f acc = {};
#pragma unroll
        for (int k = 0; k < 8; ++k) {
            int kr = 4 * k + 2 * h;
            v2f b;
            b.x = sW[(kr + 0) * D + nt * 16 + m];
            b.y = sW[(kr + 1) * D + nt * 16 + m];
            acc = __builtin_amdgcn_wmma_f32_16x16x4_f32(
                false, a[k], false, b, (short)0, acc, false, false);
        }
        int col = nt * 16 + m;
        float badd = has_bias ? bias[col] : 0.0f;   // hoisted: one load per n-tile
#pragma unroll
        for (int r = 0; r < 8; ++r) {
            int row = row0 + r + 8 * h;
            C[(size_t)row * D + col] = acc[r] + badd;
        }
    }
}

// ---------------------------------------------------------------------------
// lin2: out[nrows,32] = [A0|A1|A2][nrows,96] @ W[96,32] + bias
// ---------------------------------------------------------------------------
__global__ void gemm96_k(const float* __restrict__ A0, const float* __restrict__ A1,
                         const float* __restrict__ A2, const float* __restrict__ W,
                         const float* __restrict__ bias, float* __restrict__ C, int nrows) {
    __shared__ float sW[96 * D];
    for (int i = threadIdx.x; i < 96 * D; i += blockDim.x) sW[i] = W[i];
    __syncthreads();

    int wave = threadIdx.x >> 5;
    int lane = threadIdx.x & 31;
    int row0 = (blockIdx.x * 4 + wave) * 16;
    if (row0 >= nrows) return;

    int m = lane & 15;
    int h = lane >> 4;
    const float* bufs[3] = {A0, A1, A2};

    v8f acc0 = {}, acc1 = {};
#pragma unroll
    for (int k = 0; k < 24; ++k) {
        int kg = 4 * k + 2 * h;                 // global K (never crosses a 32 boundary inside pair)
        const float* Ab = bufs[kg >> 5];
        int kl = kg & 31;
        v2f a = *(const v2f*)(Ab + (size_t)(row0 + m) * D + kl);
        v2f b0, b1;
        b0.x = sW[(kg + 0) * D + m];       b0.y = sW[(kg + 1) * D + m];
        b1.x = sW[(kg + 0) * D + 16 + m];  b1.y = sW[(kg + 1) * D + 16 + m];
        acc0 = __builtin_amdgcn_wmma_f32_16x16x4_f32(false, a, false, b0, (short)0, acc0, false, false);
        acc1 = __builtin_amdgcn_wmma_f32_16x16x4_f32(false, a, false, b1, (short)0, acc1, false, false);
    }
    float b0v = bias[m];                        // hoisted: loop-invariant bias loads
    float b1v = bias[16 + m];
#pragma unroll
    for (int r = 0; r < 8; ++r) {
        int row = row0 + r + 8 * h;
        C[(size_t)row * D + m]      = acc0[r] + b0v;
        C[(size_t)row * D + 16 + m] = acc1[r] + b1v;
    }
}

// ---------------------------------------------------------------------------
// edge scatter: agg[dst] += xw[src] * rdeg[src]*rdeg[dst], over one bucket
// w >= 0 -> windowed bucket [bktoff[w], +bktcnt[w]); w < 0 -> all E edges
// 8 threads per edge, float4 chunks, grid-stride
// ---------------------------------------------------------------------------
__global__ void scatter_edges_k(const int* __restrict__ ebkt, const int* __restrict__ bktoff,
                                const int* __restrict__ bktcnt, int w, int Etot,
                                const int* __restrict__ src, const int* __restrict__ dst,
                                const float* __restrict__ rdeg,
                                const float* __restrict__ xw, float* __restrict__ agg) {
    int base, cnt;
    if (w < 0) { base = 0; cnt = Etot; }
    else       { base = bktoff[w]; cnt = bktcnt[w]; }
    long total  = (long)cnt * 8;
    long stride = (long)gridDim.x * blockDim.x;
    for (long t = (long)blockIdx.x * blockDim.x + threadIdx.x; t < total; t += stride) {
        long ei = t >> 3;
        int  c  = ((int)t & 7) * 4;
        if (c == 0) {                                   // prefetch edge-id stream ahead
            long pf = ei + 4096;
            if (pf < cnt) __builtin_prefetch(&ebkt[base + pf], 0, 1);
        }
        int e = ebkt[base + ei];
        int s = src[e], d = dst[e];
        float nrm = rdeg[s] * rdeg[d];
        float4 v = *(const float4*)(xw + (size_t)s * D + c);
        float* ap = agg + (size_t)d * D + c;
        atomicAdd(ap + 0, v.x * nrm);
        atomicAdd(ap + 1, v.y * nrm);
        atomicAdd(ap + 2, v.z * nrm);
        atomicAdd(ap + 3, v.w * nrm);
    }
}

// ---------------------------------------------------------------------------
// agg[n,d] += xw[n,d]*inv[n] + bias[d]  (self-loop + conv bias), then
// accumulate per-feature sum / sumsq into stats[0..31] / stats[32..63]
// ---------------------------------------------------------------------------
__global__ void finalize_reduce_k(float* __restrict__ agg, const float* __restrict__ xw,
                                  const float* __restrict__ inv, const float* __restrict__ bias,
                                  float* __restrict__ stats, int n) {
    __shared__ float s1[256], s2[256];
    int t = threadIdx.x;
    int d = t & 31;
    int r = t >> 5;                 // 8 row groups per block
    float b = bias[d];
    float sum = 0.f, sq = 0.f;
    for (int row = blockIdx.x * 8 + r; row < n; row += gridDim.x * 8) {
        size_t i = (size_t)row * D + d;
        float y = agg[i] + xw[i] * inv[row] + b;
        agg[i] = y;
        sum += y;
        sq  += y * y;
    }
    s1[t] = sum; s2[t] = sq;
    __syncthreads();
    for (int off = 4; off >= 1; off >>= 1) {
        if (r < off) { s1[t] += s1[t + off * 32]; s2[t] += s2[t + off * 32]; }
        __syncthreads();
    }
    if (r == 0) {
        atomicAdd(&stats[d],      s1[t]);
        atomicAdd(&stats[32 + d], s2[t]);
    }
}

// BatchNorm (training): y = (x - m)*rsqrt(v+eps)*g + beta, biased variance
__global__ void bn_normalize_k(const float* __restrict__ x, const float* __restrict__ stats,
                               const float* __restrict__ g, const float* __restrict__ beta,
                               float* __restrict__ out, long nelem, float invN) {
    long i = (long)blockIdx.x * blockDim.x + threadIdx.x;
    if (i >= nelem) return;
    int d = (int)(i & 31);
    float m = stats[d] * invN;
    float v = stats[32 + d] * invN - m * m;
    out[i] = (x[i] - m) * rsqrtf(v + BN_EPS) * g[d] + beta[d];
}

// ---------------------------------------------------------------------------
// host-side stage enqueue: one GCNConv + BatchNorm
// ---------------------------------------------------------------------------
static void enqueue_stage(const float* xin, const float* w, const float* cbias,
                          const float* g, const float* beta,
                          const float* rdeg, const float* inv,
                          const int* ebkt, const int* bktoff, const int* bktcnt, int win,
                          const int* src, const int* dst,
                          float* xw, float* agg /* stats immediately follows */,
                          float* stats, float* xout,
                          int N, int E, hipStream_t stream) {
    int gemm_blocks = (N + 63) / 64;
    gemm32_k<<<gemm_blocks, 128, 0, stream>>>(xin, w, nullptr, 0, xw, N);
    long zero4 = ((long)N * D + 64) / 4;                    // agg + stats contiguous
    fill_zero_k<<<(int)((zero4 + 255) / 256), 256, 0, stream>>>((uint32_t*)agg, zero4);
    scatter_edges_k<<<2048, 256, 0, stream>>>(ebkt, bktoff, bktcnt, win, E,
                                              src, dst, rdeg, xw, agg);
    finalize_reduce_k<<<512, 256, 0, stream>>>(agg, xw, inv, cbias, stats, N);
    long nelem = (long)N * D;
    bn_normalize_k<<<(int)((nelem + 255) / 256), 256, 0, stream>>>(
        agg, stats, g, beta, xout, nelem, 1.0f / (float)N);
}

extern "C" void kernel_launch(void* const* d_in, const int* in_sizes, int n_in,
                              void* d_out, int out_size, void* d_ws, size_t ws_size,
                              hipStream_t stream) {
    const int N = in_sizes[0] / D;           // 100000
    const int E = in_sizes[2];               // 1600000

    const float* x        = (const float*)d_in[0];
    const int*   ei       = (const int*)  d_in[1];
    const int*   ewin     = (const int*)  d_in[2];
    const float* lin1_w   = (const float*)d_in[3];
    const float* lin1_b   = (const float*)d_in[4];
    const float* conv_w   = (const float*)d_in[5];
    const float* conv_b   = (const float*)d_in[6];
    const float* convf_w  = (const float*)d_in[7];
    const float* convf_b  = (const float*)d_in[8];
    const float* convb_w  = (const float*)d_in[9];
    const float* convb_b  = (const float*)d_in[10];
    const float* bn_g     = (const float*)d_in[11];
    const float* bn_b     = (const float*)d_in[12];
    const float* bnf_g    = (const float*)d_in[13];
    const float* bnf_b    = (const float*)d_in[14];
    const float* bnb_g    = (const float*)d_in[15];
    const float* bnb_b    = (const float*)d_in[16];
    const float* lin2_w   = (const float*)d_in[17];
    const float* lin2_b   = (const float*)d_in[18];

    const int* src = ei;
    const int* dst = ei + E;
    float* out = (float*)d_out;

    // ---- workspace carve ----
    float* f = (float*)d_ws;
    float* h     = f; f += (size_t)N * D;
    float* cur   = f; f += (size_t)N * D;
    float* xw    = f; f += (size_t)N * D;
    float* agg   = f; f += (size_t)N * D;
    float* stats = f; f += 64;               // must follow agg (zeroed together)
    float* outa  = f; f += (size_t)N * D;
    float* outf  = f; f += (size_t)N * D;
    float* outb  = f; f += (size_t)N * D;
    float* rdegw = f; f += (size_t)NWIN * N;
    float* invw  = f; f += (size_t)NWIN * N;
    float* rdegf = f; f += (size_t)N;
    float* invf  = f; f += (size_t)N;
    int* ip     = (int*)f;
    int* cntw   = ip; ip += (size_t)NWIN * N;
    int* bktcnt = ip; ip += 8;
    int* bktoff = ip; ip += 8;
    int* cursor = ip; ip += 8;
    int* ebkt   = ip; ip += (size_t)E;

    // ---- one-time degree + edge-bucket precompute (deterministic each call) ----
    long cz4 = ((long)NWIN * N + 8) / 4;     // cntw + bktcnt contiguous, zero together
    fill_zero_k<<<(int)((cz4 + 255) / 256), 256, 0, stream>>>((uint32_t*)cntw, cz4);
    count_edges_k<<<(E + 255) / 256, 256, 0, stream>>>(dst, ewin, cntw, bktcnt, E, N);
    deg_finalize_k<<<(N + 255) / 256, 256, 0, stream>>>(cntw, rdegw, invw, rdegf, invf, N);
    bucket_prefix_k<<<1, 64, 0, stream>>>(bktcnt, bktoff, cursor);
    bucket_scatter_k<<<(E + 255) / 256, 256, 0, stream>>>(ewin, bktoff, cursor, ebkt, E);

    // ---- lin1: h = x @ W1 + b1 (WMMA) ----
    gemm32_k<<<(N + 63) / 64, 128, 0, stream>>>(x, lin1_w, lin1_b, 1, h, N);

    // ---- full-graph stream: one conv + BN -> outa ----
    enqueue_stage(h, conv_w, conv_b, bn_g, bn_b, rdegf, invf,
                  ebkt, bktoff, bktcnt, /*win=*/-1, src, dst,
                  xw, agg, stats, outa, N, E, stream);

    // ---- forward sweep: windows 0..7 -> outf ----
    const float* p = h;
    for (int i = 0; i < NWIN; ++i) {
        float* dstbuf = (i == NWIN - 1) ? outf : cur;
        enqueue_stage(p, convf_w, convf_b, bnf_g, bnf_b,
                      rdegw + (size_t)i * N, invw + (size_t)i * N,
                      ebkt, bktoff, bktcnt, i, src, dst,
                      xw, agg, stats, dstbuf, N, E, stream);
        p = dstbuf;
    }

    // ---- backward sweep: windows 7..0 -> outb ----
    p = h;
    for (int i = NWIN - 1; i >= 0; --i) {
        float* dstbuf = (i == 0) ? outb : cur;
        enqueue_stage(p, convb_w, convb_b, bnb_g, bnb_b,
                      rdegw + (size_t)i * N, invw + (size_t)i * N,
                      ebkt, bktoff, bktcnt, i, src, dst,
                      xw, agg, stats, dstbuf, N, E, stream);
        p = dstbuf;
    }

    // ---- lin2: out = [outa|outf|outb] @ W2 + b2 (WMMA) ----
    gemm96_k<<<(N + 63) / 64, 128, 0, stream>>>(outa, outf, outb, lin2_w, lin2_b, out, N);
}